// TensorNeRF_20925080666855
// MI455X (gfx1250) — compile-verified
//
#include <hip/hip_runtime.h>
#include <stdint.h>

// ---------------------------------------------------------------------------
// TensorNeRF hot path for MI455X (gfx1250, wave32).
//
// Roofline: 256 MB streamed @ 23.3 TB/s => ~11 us floor; ~50M transcendentals
// and ~0.5 GFLOP are far below compute limits => memory bound. Design:
//   * one wave (32 lanes) per ray; 16 iterations of 32 samples
//   * cumprod along S done as a wave-wide multiplicative inclusive scan
//     (shuffles -> ds permutes on CDNA5); only the chunk-carry Tc is serial
//   * sigma row (2 KB) DMA'd into LDS by the Tensor Data Mover
//     (tensor_load_to_lds + s_wait_tensorcnt); rgb streamed directly with
//     coalesced b96 loads (read-once data, no reuse => no LDS staging)
//   * loop unrolled x4 + global_prefetch_b8 two chunks ahead to maximize
//     loads-in-flight around the latency-bound scan chains
// ---------------------------------------------------------------------------

#define WAVES_PER_BLOCK 8
#define NSAMP 512

typedef __attribute__((ext_vector_type(4))) unsigned int v4u;
typedef __attribute__((ext_vector_type(4))) int          v4i;
typedef __attribute__((ext_vector_type(8))) int          v8i;

// Issue a TDM load of `nelem` 4-byte elements from contiguous global memory
// at `gaddr` into LDS byte address `lds_byte`. All args must be wave-uniform.
// D# layout per CDNA5 ISA (§8.3/8.4).
__device__ __forceinline__ void tdm_load_row_f32(uint32_t lds_byte,
                                                 uint64_t gaddr,
                                                 uint32_t nelem) {
  v4u g0;
  g0.x = 1u;                                        // count=1, user desc
  g0.y = lds_byte;                                  // lds_addr (bytes)
  g0.z = (uint32_t)gaddr;                           // global_addr[31:0]
  g0.w = ((uint32_t)(gaddr >> 32) & 0x01FFFFFFu)    // global_addr[56:32]
         | 0x80000000u;                             // type = 2 ("image")

  const uint32_t td0 = nelem;    // tensor_dim0 (data_size units)
  const uint32_t td1 = 1u;       // tensor_dim1
  const uint32_t tile0 = nelem;  // tile_dim0 (16-bit, 512 fits)
  v8i g1;
  g1[0] = (int)(2u << 16);                               // data_size=2 (4B), mask=0
  g1[1] = (int)((td0 & 0xFFFFu) << 16);                  // tensor_dim0[15:0]
  g1[2] = (int)((td0 >> 16) | ((td1 & 0xFFFFu) << 16));  // td0[31:16] | td1[15:0]
  g1[3] = (int)((td1 >> 16) | (tile0 << 16));            // td1[31:16] | tile_dim0
  g1[4] = 0;                                             // tile_dim1=0, tile_dim2=0
  g1[5] = (int)td0;                                      // tensor_dim0_stride[31:0]
  g1[6] = 0;                                             // stride hi / dim1_stride lo
  g1[7] = 0;
  v4i gz = {0, 0, 0, 0};
#if defined(__clang_major__) && (__clang_major__ >= 23)
  v8i gz8 = {0, 0, 0, 0, 0, 0, 0, 0};
  __builtin_amdgcn_tensor_load_to_lds(g0, g1, gz, gz, gz8, 0);
#else
  __builtin_amdgcn_tensor_load_to_lds(g0, g1, gz, gz, 0);
#endif
}

__global__ __launch_bounds__(WAVES_PER_BLOCK * 32)
void nerf_composite_kernel(const float* __restrict__ rays_o,
                           const float* __restrict__ rays_d,
                           const float* __restrict__ sigma_feat,
                           const float* __restrict__ rgb,
                           const float* __restrict__ aabb,
                           float* __restrict__ out,
                           int N) {
  __shared__ float s_sigma[WAVES_PER_BLOCK][NSAMP];

  const int lane = threadIdx.x & 31;
  const int wid  = threadIdx.x >> 5;
  const uint32_t n =
      (uint32_t)__builtin_amdgcn_readfirstlane(blockIdx.x * WAVES_PER_BLOCK + wid);
  if ((int)n >= N) return;

  // ---- TDM: DMA this wave's sigma row (2 KB) into LDS --------------------
  const uint32_t lds_off = (uint32_t)__builtin_amdgcn_readfirstlane(
      (int)(uint32_t)(uintptr_t)&s_sigma[wid][0]);
  const uint64_t ga = (uint64_t)(uintptr_t)(sigma_feat + (size_t)n * NSAMP);
  const uint32_t ga_lo = (uint32_t)__builtin_amdgcn_readfirstlane((int)(uint32_t)ga);
  const uint32_t ga_hi = (uint32_t)__builtin_amdgcn_readfirstlane((int)(uint32_t)(ga >> 32));
  tdm_load_row_f32(lds_off, ((uint64_t)ga_hi << 32) | ga_lo, NSAMP);

  // ---- per-ray uniforms (scalar loads; n is wave-uniform) ----------------
  const float ox = rays_o[3 * n + 0], oy = rays_o[3 * n + 1], oz = rays_o[3 * n + 2];
  const float dx = rays_d[3 * n + 0], dy = rays_d[3 * n + 1], dz = rays_d[3 * n + 2];
  const float lox = aabb[0], loy = aabb[1], loz = aabb[2];
  const float hix = aabb[3], hiy = aabb[4], hiz = aabb[5];

  const float NEARv = 2.0f, FARv = 6.0f;
  const float dt    = (FARv - NEARv) / (float)(NSAMP - 1);  // linspace step
  const float dist  = dt * 25.0f;                           // DISTANCE_SCALE

  const float* __restrict__ rgb_row = rgb + (size_t)n * NSAMP * 3;

  // warm up the rgb stream while the TDM is in flight
  __builtin_prefetch(rgb_row + 3 * lane, 0, 0);
  __builtin_prefetch(rgb_row + 3 * (32 + lane), 0, 0);

  float accR = 0.0f, accG = 0.0f, accB = 0.0f, accD = 0.0f;
  float Tc = 1.0f;  // running transmittance carried across 32-sample chunks

  __builtin_amdgcn_s_wait_tensorcnt(0);
  asm volatile("" ::: "memory");  // keep LDS reads after the TDM wait

#pragma unroll 4
  for (int s0 = 0; s0 < NSAMP; s0 += 32) {
    const int s = s0 + lane;
    const float tt = NEARv + dt * (float)s;        // interpx[s]

    // stream-ahead prefetch (speculative; OOB at row end silently dropped)
    __builtin_prefetch(rgb_row + 3 * (s + 64), 0, 0);

    const float feat = s_sigma[wid][s];

    // in-AABB test
    const float px = fmaf(dx, tt, ox);
    const float py = fmaf(dy, tt, oy);
    const float pz = fmaf(dz, tt, oz);
    const bool inb = (px >= lox) & (px <= hix) & (py >= loy) & (py <= hiy) &
                     (pz >= loz) & (pz <= hiz);

    // softplus(feat - 10) = max(x,0) + log1p(exp(-|x|))
    const float x  = feat - 10.0f;
    const float sp = fmaxf(x, 0.0f) + log1pf(__expf(-fabsf(x)));
    const float sigma = inb ? sp : 0.0f;

    const float alpha = 1.0f - __expf(-sigma * dist);
    const float t     = 1.0f - alpha + 1e-10f;

    // wave-wide inclusive multiplicative scan of t (Hillis–Steele)
    float scan = t;
#pragma unroll
    for (int off = 1; off < 32; off <<= 1) {
      const float up = __shfl_up(scan, (unsigned)off, 32);
      if (lane >= off) scan *= up;
    }
    // exclusive scan via shift; lane 0 gets identity
    float excl = __shfl_up(scan, 1u, 32);
    if (lane == 0) excl = 1.0f;

    const float w = alpha * (Tc * excl);

    const float3 c = *(const float3*)(rgb_row + 3 * s);  // coalesced b96
    accR = fmaf(w, c.x, accR);
    accG = fmaf(w, c.y, accG);
    accB = fmaf(w, c.z, accB);
    accD = fmaf(w, tt, accD);

    Tc *= __shfl(scan, 31, 32);  // fold chunk product into the carry
  }

  // wave reduction of the four accumulators
#pragma unroll
  for (int off = 16; off >= 1; off >>= 1) {
    accR += __shfl_down(accR, (unsigned)off, 32);
    accG += __shfl_down(accG, (unsigned)off, 32);
    accB += __shfl_down(accB, (unsigned)off, 32);
    accD += __shfl_down(accD, (unsigned)off, 32);
  }

  if (lane == 0) {
    out[3 * n + 0] = accR;                    // rgb_map
    out[3 * n + 1] = accG;
    out[3 * n + 2] = accB;
    out[(size_t)3 * N + n] = accD;            // depth_map
    out[(size_t)4 * N + n] = Tc;              // bg_weight
  }
}

extern "C" void kernel_launch(void* const* d_in, const int* in_sizes, int n_in,
                              void* d_out, int out_size, void* d_ws, size_t ws_size,
                              hipStream_t stream) {
  const float* rays_o     = (const float*)d_in[0];
  const float* rays_d     = (const float*)d_in[1];
  const float* sigma_feat = (const float*)d_in[2];
  const float* rgb        = (const float*)d_in[3];
  const float* aabb       = (const float*)d_in[4];
  float* out = (float*)d_out;

  const int N = in_sizes[0] / 3;  // rays_o is (N, 3)

  const dim3 block(WAVES_PER_BLOCK * 32);
  const dim3 grid((N + WAVES_PER_BLOCK - 1) / WAVES_PER_BLOCK);
  nerf_composite_kernel<<<grid, block, 0, stream>>>(rays_o, rays_d, sigma_feat,
                                                    rgb, aabb, out, N);
}